// SpatialGCNModel_13795434955218
// MI455X (gfx1250) — compile-verified
//
#include <hip/hip_runtime.h>
#include <math.h>

// CDNA5 / gfx1250. wave32.
typedef __attribute__((ext_vector_type(2))) float v2f;
typedef __attribute__((ext_vector_type(8))) float v8f;

#define HID   16
#define HC    256   // HID * IN_CH
#define XDIM  18    // 2 pos + 16 feat

// ---------------------------------------------------------------- zero agg
__global__ void zero_kernel(float* __restrict__ p, long long n) {
    long long i = (long long)blockIdx.x * blockDim.x + threadIdx.x;
    long long stride = (long long)gridDim.x * blockDim.x;
    for (; i < n; i += stride) p[i] = 0.0f;
}

// ---------------------------------------------------------------- edge stage
// m[e,h,c] = relu(rel @ W_in + b_in)[h*16+c] * feat[src][c], scatter-add to agg[dst]
// 256 threads/block, 8 threads per edge, 32 channels per thread.
__global__ void conv_edge_kernel(const float* __restrict__ x,      // N x 18 (pos in cols 0,1)
                                 const float* __restrict__ feat,   // N x fstride
                                 int fstride,
                                 const int*   __restrict__ ei,     // 2 x E flat
                                 int E, int N,
                                 const float* __restrict__ Win,    // 2 x 256
                                 const float* __restrict__ bin,    // 256
                                 float*       __restrict__ agg)    // N x 256
{
    __shared__ float sW[2 * HC];
    __shared__ float sB[HC];
    int tid = threadIdx.x;
    for (int i = tid; i < 2 * HC; i += 256) sW[i] = Win[i];
    sB[tid] = bin[tid];
    __syncthreads();

    long long NE = (long long)E + N;                 // edges + self loops
    long long e  = (long long)blockIdx.x * 32 + (tid >> 3);
    if (e >= NE) return;

    int s, d;
    if (e < E) { s = ei[e]; d = ei[(long long)E + e]; }
    else       { s = d = (int)(e - E); }

    float rx = x[(size_t)s * XDIM + 0] - x[(size_t)d * XDIM + 0];
    float ry = x[(size_t)s * XDIM + 1] - x[(size_t)d * XDIM + 1];

    const float* xj = feat + (size_t)s * fstride;
    float xv[16];
#pragma unroll
    for (int c = 0; c < 16; ++c) xv[c] = xj[c];

    float* outrow = agg + (size_t)d * HC;
    int chb = (tid & 7) * 32;
#pragma unroll
    for (int i = 0; i < 32; ++i) {
        int hc = chb + i;
        float sv = fmaf(rx, sW[hc], fmaf(ry, sW[HC + hc], sB[hc]));
        sv = fmaxf(sv, 0.0f);
        unsafeAtomicAdd(&outrow[hc], sv * xv[hc & 15]);   // global_atomic_add_f32, no return
    }
}

// ---------------------------------------------------------------- node GEMM (WMMA f32)
// h[N x 16] = agg[N x 256] @ Wout[256 x 16] + bout
// One wave per 16-row node tile; 64 x V_WMMA_F32_16X16X4_F32 along K=256.
__global__ void conv_node_wmma(const float* __restrict__ agg,   // N x 256
                               const float* __restrict__ Wout,  // 256 x 16 row-major
                               const float* __restrict__ bout,  // 16
                               float*       __restrict__ h,     // N x 16
                               int N, int ntiles)
{
    int wave = (int)((blockIdx.x * blockDim.x + threadIdx.x) >> 5);
    int lane = threadIdx.x & 31;
    if (wave >= ntiles) return;                      // wave-uniform: EXEC stays all-ones

    int m    = lane & 15;                            // A row / B col / D col
    int kh   = (lane >> 4) * 2;                      // lanes 16-31 hold K=2,3 of each step

    int  arow_i = wave * 16 + m;
    if (arow_i >= N) arow_i = N - 1;                 // clamp (uniform EXEC preserved)
    const float* arow = agg  + (size_t)arow_i * HC + kh;
    const float* bcol = Wout + (size_t)kh * HID + m;

    v8f acc = {};
#pragma unroll
    for (int k = 0; k < HC; k += 4) {
        v2f a, b;
        a[0] = arow[k];            a[1] = arow[k + 1];
        b[0] = bcol[k * HID];      b[1] = bcol[(k + 1) * HID];
        // D = A(16x4,f32) * B(4x16,f32) + C ; 8 args: neg_a,A,neg_b,B,c_mod,C,reuse_a,reuse_b
        acc = __builtin_amdgcn_wmma_f32_16x16x4_f32(false, a, false, b,
                                                    (short)0, acc, false, false);
    }

    float bias  = bout[m];
    int   rbase = wave * 16 + (lane >> 4) * 8;       // lanes 16-31 hold rows 8..15
#pragma unroll
    for (int r = 0; r < 8; ++r) {
        int row = rbase + r;
        if (row < N) h[(size_t)row * HID + m] = acc[r] + bias;
    }
}

// ---------------------------------------------------------------- edge decoder MLP
// z = [h[src], h[dst]] (32) -> relu(·Wd1+b) -> relu(·Wd2+b) -> sigmoid(·Wd3+b)
__global__ void decode_kernel(const float* __restrict__ h,   // N x 16
                              const int*   __restrict__ ei,  // 2 x E flat
                              int E,
                              const float* __restrict__ Wd1, const float* __restrict__ bd1,
                              const float* __restrict__ Wd2, const float* __restrict__ bd2,
                              const float* __restrict__ Wd3, const float* __restrict__ bd3,
                              float*       __restrict__ out)
{
    __shared__ float sW1[32 * 16];
    __shared__ float sW2[16 * 16];
    __shared__ float sW3[16];
    __shared__ float sb1[16];
    __shared__ float sb2[16];
    int tid = threadIdx.x;
    for (int i = tid; i < 512; i += 256) sW1[i] = Wd1[i];
    sW2[tid] = Wd2[tid];
    if (tid < 16) { sW3[tid] = Wd3[tid]; sb1[tid] = bd1[tid]; sb2[tid] = bd2[tid]; }
    __syncthreads();

    int e = blockIdx.x * 256 + tid;
    if (e >= E) return;

    int s = ei[e], d = ei[(long long)E + e];
    float z[32];
#pragma unroll
    for (int c = 0; c < 16; ++c) {
        z[c]      = h[(size_t)s * HID + c];
        z[16 + c] = h[(size_t)d * HID + c];
    }

    float z1[16];
#pragma unroll
    for (int j = 0; j < 16; ++j) {
        float acc = sb1[j];
#pragma unroll
        for (int i = 0; i < 32; ++i) acc = fmaf(z[i], sW1[i * 16 + j], acc);
        z1[j] = fmaxf(acc, 0.0f);
    }

    float z2[16];
#pragma unroll
    for (int j = 0; j < 16; ++j) {
        float acc = sb2[j];
#pragma unroll
        for (int i = 0; i < 16; ++i) acc = fmaf(z1[i], sW2[i * 16 + j], acc);
        z2[j] = fmaxf(acc, 0.0f);
    }

    float t = bd3[0];
#pragma unroll
    for (int i = 0; i < 16; ++i) t = fmaf(z2[i], sW3[i], t);

    out[e] = 1.0f / (1.0f + __expf(-t));
}

// ---------------------------------------------------------------- driver
extern "C" void kernel_launch(void* const* d_in, const int* in_sizes, int n_in,
                              void* d_out, int out_size, void* d_ws, size_t ws_size,
                              hipStream_t stream)
{
    const float* x    = (const float*)d_in[0];
    const int*   ei   = (const int*)  d_in[1];
    const float* Win[3]  = { (const float*)d_in[2],  (const float*)d_in[6],  (const float*)d_in[10] };
    const float* bin[3]  = { (const float*)d_in[3],  (const float*)d_in[7],  (const float*)d_in[11] };
    const float* Wout[3] = { (const float*)d_in[4],  (const float*)d_in[8],  (const float*)d_in[12] };
    const float* bout[3] = { (const float*)d_in[5],  (const float*)d_in[9],  (const float*)d_in[13] };
    const float* Wd1 = (const float*)d_in[14]; const float* bd1 = (const float*)d_in[15];
    const float* Wd2 = (const float*)d_in[16]; const float* bd2 = (const float*)d_in[17];
    const float* Wd3 = (const float*)d_in[18]; const float* bd3 = (const float*)d_in[19];
    float* out = (float*)d_out;

    const int N = in_sizes[0] / XDIM;
    const int E = in_sizes[1] / 2;
    const long long NE = (long long)E + N;

    float* agg = (float*)d_ws;                       // N*256 f32
    float* h1  = agg + (size_t)N * HC;               // N*16
    float* h2  = h1  + (size_t)N * HID;
    float* h3  = h2  + (size_t)N * HID;
    float* hbuf[3] = { h1, h2, h3 };

    const long long aggN   = (long long)N * HC;
    const int ntiles       = (N + 15) / 16;
    const int edgeBlocks   = (int)((NE + 31) / 32);
    const int nodeBlocks   = (ntiles + 7) / 8;       // 8 waves / 256-thread block

    const float* feat = x + 2;  int fstride = XDIM;  // layer 1: features live inside x
    for (int l = 0; l < 3; ++l) {
        zero_kernel<<<2048, 256, 0, stream>>>(agg, aggN);
        conv_edge_kernel<<<edgeBlocks, 256, 0, stream>>>(x, feat, fstride, ei, E, N,
                                                         Win[l], bin[l], agg);
        conv_node_wmma<<<nodeBlocks, 256, 0, stream>>>(agg, Wout[l], bout[l],
                                                       hbuf[l], N, ntiles);
        feat = hbuf[l]; fstride = HID;               // later layers read previous h
    }

    decode_kernel<<<(E + 255) / 256, 256, 0, stream>>>(h3, ei, E,
                                                       Wd1, bd1, Wd2, bd2, Wd3, bd3, out);
}